// GALA_65146063945780
// MI455X (gfx1250) — compile-verified
//
#include <hip/hip_runtime.h>

typedef __attribute__((ext_vector_type(2))) float v2f;
typedef __attribute__((ext_vector_type(8))) float v8f;

// ---------------------------------------------------------------------------
// Degree / normalization
// ---------------------------------------------------------------------------
__global__ void gala_zero_f32(float* __restrict__ p, int n) {
  int i = blockIdx.x * blockDim.x + threadIdx.x;
  int stride = gridDim.x * blockDim.x;
  for (; i < n; i += stride) p[i] = 0.0f;
}

__global__ void gala_deg_count(const int* __restrict__ ei, float* __restrict__ deg, int E) {
  int i = blockIdx.x * blockDim.x + threadIdx.x;
  int stride = gridDim.x * blockDim.x;
  for (; i < E; i += stride) atomicAdd(&deg[ei[E + i]], 1.0f);
}

__global__ void gala_calc_dis(float* __restrict__ degdis, int n) {
  int i = blockIdx.x * blockDim.x + threadIdx.x;
  int stride = gridDim.x * blockDim.x;
  for (; i < n; i += stride) degdis[i] = rsqrtf(degdis[i] + 1.0f);
}

// ---------------------------------------------------------------------------
// Dense GEMM: D[N x dout] = A[N x din] * W[din x dout], exact fp32 via
// V_WMMA_F32_16X16X4_F32. One wave per 16x16 output tile, K stepped by 4.
//
// Lane layouts per CDNA5 ISA 7.12.2:
//   A 16x4 : lane M = lane&15 ; v0 = K0 (lanes 0-15) / K2 (lanes 16-31), v1 = K1/K3
//   B 4x16 : lane N = lane&15 ; v0 = K0/K2 row, v1 = K1/K3 row
//   C 16x16: lane N = lane&15 ; vr = row r (lanes 0-15) / row r+8 (lanes 16-31)
// ---------------------------------------------------------------------------
__global__ void gala_gemm_wmma(const float* __restrict__ A,
                               const float* __restrict__ W,
                               float* __restrict__ D,
                               int rowTiles, int din, int dout) {
  const int lane    = threadIdx.x & 31;
  const int lane15  = lane & 15;
  const int half    = lane >> 4;             // 0: lanes 0-15, 1: lanes 16-31
  const int wavesPerBlock = blockDim.x >> 5;
  const int wave    = blockIdx.x * wavesPerBlock + (threadIdx.x >> 5);
  const int numWaves = gridDim.x * wavesPerBlock;
  const int colTiles = dout >> 4;
  const int totalTiles = rowTiles * colTiles;

  for (int tile = wave; tile < totalTiles; tile += numWaves) {
    const int rt = tile / colTiles;
    const int ct = tile - rt * colTiles;
    const int row0 = rt << 4;
    const int col0 = ct << 4;

    v8f acc = {};
    const float* __restrict__ arow = A + (size_t)(row0 + lane15) * din;
    for (int k = 0; k < din; k += 4) {
      const int ka = k + (half << 1);
      v2f a;
      a.x = arow[ka];
      a.y = arow[ka + 1];
      v2f b;
      b.x = W[(size_t)ka * dout + col0 + lane15];
      b.y = W[(size_t)(ka + 1) * dout + col0 + lane15];
      acc = __builtin_amdgcn_wmma_f32_16x16x4_f32(
          /*neg_a=*/false, a, /*neg_b=*/false, b,
          /*c_mod=*/(short)0, acc, /*reuse_a=*/false, /*reuse_b=*/false);
    }

    float* __restrict__ dptr = D + (size_t)(row0 + (half << 3)) * dout + col0 + lane15;
#pragma unroll
    for (int r = 0; r < 8; ++r) {
      dptr[(size_t)r * dout] = acc[r];
    }
  }
}

// ---------------------------------------------------------------------------
// agg[i,c] = hW[i,c] * dis[i]^2 + b[c]   (self-loop + bias, also zero-inits agg)
// dout is a power of two: pass shift.
// ---------------------------------------------------------------------------
__global__ void gala_init_agg(const float* __restrict__ hw,
                              const float* __restrict__ dis,
                              const float* __restrict__ bias,
                              float* __restrict__ agg,
                              int n, int doutShift) {
  const int mask = (1 << doutShift) - 1;
  int idx = blockIdx.x * blockDim.x + threadIdx.x;
  int stride = gridDim.x * blockDim.x;
  for (; idx < n; idx += stride) {
    const int i = idx >> doutShift;
    const int c = idx & mask;
    const float d = dis[i];
    agg[idx] = hw[idx] * (d * d) + bias[c];
  }
}

// ---------------------------------------------------------------------------
// Edge scatter: agg[dst, :] += hW[src, :] * dis[src] * dis[dst]
// One wave per edge; lanes striped across channels (coalesced gather + atomics).
// ---------------------------------------------------------------------------
__global__ void gala_scatter(const int* __restrict__ ei,
                             const float* __restrict__ hw,
                             const float* __restrict__ dis,
                             float* __restrict__ agg,
                             int E, int dout) {
  const int lane = threadIdx.x & 31;
  const int wave = (blockIdx.x * blockDim.x + threadIdx.x) >> 5;
  const int numWaves = (gridDim.x * blockDim.x) >> 5;
  for (int e = wave; e < E; e += numWaves) {
    const int s = ei[e];
    const int d = ei[E + e];
    const float w = dis[s] * dis[d];
    const float* __restrict__ srow = hw + (size_t)s * dout;
    float* __restrict__ drow = agg + (size_t)d * dout;
    for (int c = lane; c < dout; c += 32) {
      atomicAdd(&drow[c], srow[c] * w);
    }
  }
}

// ---------------------------------------------------------------------------
// h_out = relu?(agg); optionally duplicate into out2 (for emitting z)
// ---------------------------------------------------------------------------
__global__ void gala_finish(const float* __restrict__ agg,
                            float* __restrict__ out,
                            float* __restrict__ out2,
                            int n, int doRelu) {
  int idx = blockIdx.x * blockDim.x + threadIdx.x;
  int stride = gridDim.x * blockDim.x;
  for (; idx < n; idx += stride) {
    float v = agg[idx];
    if (doRelu) v = v > 0.0f ? v : 0.0f;
    out[idx] = v;
    if (out2) out2[idx] = v;
  }
}

// ---------------------------------------------------------------------------
// Orchestration
// ---------------------------------------------------------------------------
extern "C" void kernel_launch(void* const* d_in, const int* in_sizes, int n_in,
                              void* d_out, int out_size, void* d_ws, size_t ws_size,
                              hipStream_t stream) {
  (void)in_sizes; (void)n_in; (void)out_size; (void)ws_size;

  constexpr int N  = 100000;
  constexpr int E  = 1600000;
  constexpr int IN = 128;

  const float* x  = (const float*)d_in[0];
  const int*   ei = (const int*)d_in[1];

  const float* Ws[6] = { (const float*)d_in[2], (const float*)d_in[3], (const float*)d_in[4],
                         (const float*)d_in[8], (const float*)d_in[9], (const float*)d_in[10] };
  const float* bs[6] = { (const float*)d_in[5], (const float*)d_in[6], (const float*)d_in[7],
                         (const float*)d_in[11], (const float*)d_in[12], (const float*)d_in[13] };
  const int dins[6]   = {128, 128, 64, 32, 64, 128};
  const int douts[6]  = {128, 64, 32, 64, 128, 128};
  const int dshift[6] = {7, 6, 5, 6, 7, 7};

  float* out = (float*)d_out;              // x_hat: [N,128] at offset 0
  float* zOut = out + (size_t)N * IN;      // z:     [N,32]  after x_hat

  // Workspace layout
  char* ws = (char*)d_ws;
  size_t off = 0;
  auto alloc = [&](size_t bytes) -> float* {
    float* p = (float*)(ws + off);
    off += (bytes + 255) & ~size_t(255);
    return p;
  };
  float* dis    = alloc((size_t)N * sizeof(float));        // deg -> dis in place
  float* bufH   = alloc((size_t)N * IN * sizeof(float));   // layer inputs/outputs
  float* bufHW  = alloc((size_t)N * IN * sizeof(float));   // h @ W
  float* bufAgg = alloc((size_t)N * IN * sizeof(float));   // aggregation

  const int BLK = 256;
  const int gridN  = (N + BLK - 1) / BLK;
  const int gridE  = (E + BLK - 1) / BLK;

  // --- normalization coefficients ---
  gala_zero_f32<<<gridN, BLK, 0, stream>>>(dis, N);
  gala_deg_count<<<gridE, BLK, 0, stream>>>(ei, dis, E);
  gala_calc_dis<<<gridN, BLK, 0, stream>>>(dis, N);

  // --- 6 GCN layers ---
  const float* h = x;
  const int rowTiles = N / 16;
  const int wavesPerBlock = BLK / 32;

  for (int l = 0; l < 6; ++l) {
    const int din = dins[l], dout = douts[l];

    // 1) hW = h @ W   (fp32 WMMA)
    {
      const int totalTiles = rowTiles * (dout >> 4);
      const int grid = (totalTiles + wavesPerBlock - 1) / wavesPerBlock;
      gala_gemm_wmma<<<grid, BLK, 0, stream>>>(h, Ws[l], bufHW, rowTiles, din, dout);
    }

    // 2) agg = hW * self_norm + b
    {
      const int n = N * dout;
      const int grid = (n + BLK - 1) / BLK;
      gala_init_agg<<<grid, BLK, 0, stream>>>(bufHW, dis, bs[l], bufAgg, n, dshift[l]);
    }

    // 3) agg[dst] += hW[src] * norm(e)
    {
      int grid = (E + wavesPerBlock - 1) / wavesPerBlock;
      if (grid > 131072) grid = 131072;   // grid-stride handles the rest
      gala_scatter<<<grid, BLK, 0, stream>>>(ei, bufHW, dis, bufAgg, E, dout);
    }

    // 4) activation / output routing
    {
      const int n = N * dout;
      const int grid = (n + BLK - 1) / BLK;
      if (l == 5) {
        // final decoder layer: x_hat, no ReLU, straight to d_out
        gala_finish<<<grid, BLK, 0, stream>>>(bufAgg, out, nullptr, n, 0);
      } else {
        float* dup = (l == 2) ? zOut : nullptr;  // latent z after encoder
        gala_finish<<<grid, BLK, 0, stream>>>(bufAgg, bufH, dup, n, 1);
        h = bufH;
      }
    }
  }
}